// mdn_77335181131795
// MI455X (gfx1250) — compile-verified
//
#include <hip/hip_runtime.h>
#include <math.h>

typedef __attribute__((ext_vector_type(2))) float v2f;
typedef __attribute__((ext_vector_type(8))) float v8f;

#define N_TOT 524288
#define DX 128
#define DT 64
#define MM 32
#define NK 8

__device__ __forceinline__ float wave_half16_sum(float v) {
    v += __shfl_xor(v, 1, 16);
    v += __shfl_xor(v, 2, 16);
    v += __shfl_xor(v, 4, 16);
    v += __shfl_xor(v, 8, 16);
    return v;
}

// Low 32 bits of a generic pointer to LDS == LDS byte offset (aperture layout).
__device__ __forceinline__ unsigned lds_addr_of(const void* p) {
    return (unsigned)(unsigned long long)(uintptr_t)p;
}

__global__ void mdn_zero(float* out) { out[0] = 0.0f; }

// Regularizer: SK*(|Wm|^2+|Wv|^2) + SS*|Wg|^2 with SK=SS=1
__global__ __launch_bounds__(256) void mdn_reg(const float* __restrict__ Wm,
                                               const float* __restrict__ Wv,
                                               const float* __restrict__ Wg,
                                               float* __restrict__ out) {
    const int tid = threadIdx.x;
    float acc = 0.0f;
    for (int i = tid; i < NK * DX * MM; i += 256) {
        float a = Wm[i]; float b = Wv[i];
        acc = fmaf(a, a, acc);
        acc = fmaf(b, b, acc);
    }
    for (int i = tid; i < DT * NK; i += 256) {
        float g = Wg[i];
        acc = fmaf(g, g, acc);
    }
    acc += __shfl_xor(acc, 1, 32);
    acc += __shfl_xor(acc, 2, 32);
    acc += __shfl_xor(acc, 4, 32);
    acc += __shfl_xor(acc, 8, 32);
    acc += __shfl_xor(acc, 16, 32);
    __shared__ float ws[8];
    if ((tid & 31) == 0) ws[tid >> 5] = acc;
    __syncthreads();
    if (tid == 0) {
        float s = 0.0f;
        #pragma unroll
        for (int i = 0; i < 8; ++i) s += ws[i];
        atomicAdd(out, s);
    }
}

__global__ __launch_bounds__(256) void mdn_main(
    const float* __restrict__ x, const float* __restrict__ t,
    const float* __restrict__ y, const float* __restrict__ Wm,
    const float* __restrict__ bm, const float* __restrict__ Wv,
    const float* __restrict__ bv, const float* __restrict__ Wg,
    const float* __restrict__ bg, float* __restrict__ out) {
    // Weight table: wlds[d/2][col*2 + (d&1)], col = (is_v?256:0) + k*32 + m. 256 KB.
    __shared__ float wlds[64][1024];
    __shared__ float wg_s[DT * NK];
    __shared__ float bg_s[NK];
    __shared__ float pi_s[8][16][NK];

    const int tid = threadIdx.x;
    const int lane = tid & 31;
    const int wv = tid >> 5;
    const int l15 = lane & 15;
    const int hi = lane >> 4;  // 0 or 1
    const float HALF_LOG_2PI = 0.91893853320467274178f;

    // ---- Stage Wm||Wv into LDS via async memory->LDS copies (ASYNCcnt) ----
    // Per-lane B32 scatter is required because the LDS layout interleaves
    // d-parity pairs for single-ds_load_b64 WMMA B fragments.
    for (int d = 0; d < DX; ++d) {
        const int dp = d >> 1, jb = d & 1;
        #pragma unroll
        for (int c0 = 0; c0 < 512; c0 += 256) {
            const int col = c0 + tid;
            const int k = (col >> 5) & 7;
            const int m = col & 31;
            const float* src = (col < 256) ? Wm : Wv;
            const unsigned long long gaddr =
                (unsigned long long)(uintptr_t)&src[(k * DX + d) * MM + m];
            const unsigned laddr = lds_addr_of(&wlds[dp][col * 2 + jb]);
            asm volatile("global_load_async_to_lds_b32 %0, %1, off"
                         :: "v"(laddr), "v"(gaddr)
                         : "memory");
        }
    }
    if (tid < DT * NK) wg_s[tid] = Wg[tid];
    if (tid < NK) bg_s[tid] = bg[tid];
    // Each wave drains its own async copies before the block barrier.
    asm volatile("s_wait_asynccnt 0" ::: "memory");
    __syncthreads();

    const int row0 = (blockIdx.x * 8 + wv) * 16;

    // ---- A fragments: 16x128 x-tile in WMMA f32 16x16x4 A layout ----
    v2f a[32];
    {
        const float* xr = x + (size_t)(row0 + l15) * DX + (hi << 1);
        #pragma unroll
        for (int s = 0; s < 32; ++s) a[s] = *(const v2f*)(xr + 4 * s);
    }

    // ---- y fragments in C layout (vgpr e -> sample row e + 8*hi) ----
    float yf[2][8];
    {
        const int srow = row0 + (hi << 3);
        #pragma unroll
        for (int e = 0; e < 8; ++e) {
            yf[0][e] = y[(size_t)(srow + e) * MM + l15];
            yf[1][e] = y[(size_t)(srow + e) * MM + 16 + l15];
        }
    }

    // ---- Gate softmax for sample l15 (duplicated across halves) ----
    {
        float g[NK];
        #pragma unroll
        for (int k = 0; k < NK; ++k) g[k] = bg_s[k];
        const float* tr = t + (size_t)(row0 + l15) * DT;
        for (int d = 0; d < DT; ++d) {
            const float td = tr[d];
            #pragma unroll
            for (int k = 0; k < NK; ++k) g[k] = fmaf(td, wg_s[d * NK + k], g[k]);
        }
        float mx = g[0];
        #pragma unroll
        for (int k = 1; k < NK; ++k) mx = fmaxf(mx, g[k]);
        float ssum = 0.0f, ex[NK];
        #pragma unroll
        for (int k = 0; k < NK; ++k) { ex[k] = __expf(g[k] - mx); ssum += ex[k]; }
        const float inv = 1.0f / ssum;
        if (lane < 16) {
            #pragma unroll
            for (int k = 0; k < NK; ++k) pi_s[wv][l15][k] = ex[k] * inv;
        }
    }
    // Same-wave cross-lane LDS dependency: wait on DScnt explicitly.
    asm volatile("s_wait_dscnt 0" ::: "memory");

    float mixv[8];
    #pragma unroll
    for (int e = 0; e < 8; ++e) mixv[e] = 0.0f;

    #pragma unroll 1
    for (int k = 0; k < NK; ++k) {
        float ce[8];
        #pragma unroll
        for (int e = 0; e < 8; ++e) ce[e] = 0.0f;

        #pragma unroll
        for (int mh = 0; mh < 2; ++mh) {
            const int cbm = k * MM + mh * 16;       // mean columns
            const int cbv = 256 + cbm;              // logvar columns
            v8f cm = {};
            v8f cv = {};
            #pragma unroll
            for (int s = 0; s < 32; ++s) {
                const int dp = 2 * s + hi;
                const v2f bmf = *(const v2f*)&wlds[dp][(cbm + l15) * 2];
                const v2f bvf = *(const v2f*)&wlds[dp][(cbv + l15) * 2];
                cm = __builtin_amdgcn_wmma_f32_16x16x4_f32(
                    false, a[s], false, bmf, (short)0, cm, false, false);
                cv = __builtin_amdgcn_wmma_f32_16x16x4_f32(
                    false, a[s], false, bvf, (short)0, cv, false, false);
            }
            const float bmb = bm[k * MM + mh * 16 + l15];
            const float bvb = bv[k * MM + mh * 16 + l15];
            #pragma unroll
            for (int e = 0; e < 8; ++e) {
                const float mean = cm[e] + bmb;
                const float lv = cv[e] + bvb;
                const float d = yf[mh][e] - mean;
                ce[e] += fmaf(d * d * 0.5f, __expf(-lv),
                              HALF_LOG_2PI + 0.5f * lv);
            }
        }
        #pragma unroll
        for (int e = 0; e < 8; ++e) {
            const float red = wave_half16_sum(ce[e]);   // sum over 32 features
            const float ll = -red;                      // per-sample per-expert LL
            const float pik = pi_s[wv][e + (hi << 3)][k];
            mixv[e] = fmaf(pik, __expf(ll), mixv[e]);
        }
    }

    // -sum(log(mix)) over this wave's 16 samples.
    float tsum = 0.0f;
    #pragma unroll
    for (int e = 0; e < 8; ++e) tsum += __logf(mixv[e]);
    tsum *= (1.0f / 16.0f);  // each half's value replicated across 16 lanes
    tsum += __shfl_xor(tsum, 1, 32);
    tsum += __shfl_xor(tsum, 2, 32);
    tsum += __shfl_xor(tsum, 4, 32);
    tsum += __shfl_xor(tsum, 8, 32);
    tsum += __shfl_xor(tsum, 16, 32);
    if (lane == 0) atomicAdd(out, -tsum);
}

extern "C" void kernel_launch(void* const* d_in, const int* in_sizes, int n_in,
                              void* d_out, int out_size, void* d_ws, size_t ws_size,
                              hipStream_t stream) {
    const float* x  = (const float*)d_in[0];
    const float* t  = (const float*)d_in[1];
    const float* y  = (const float*)d_in[2];
    const float* Wm = (const float*)d_in[3];
    const float* bm = (const float*)d_in[4];
    const float* Wv = (const float*)d_in[5];
    const float* bv = (const float*)d_in[6];
    const float* Wg = (const float*)d_in[7];
    const float* bg = (const float*)d_in[8];
    float* out = (float*)d_out;

    mdn_zero<<<1, 1, 0, stream>>>(out);
    mdn_main<<<N_TOT / 128, 256, 0, stream>>>(x, t, y, Wm, bm, Wv, bv, Wg, bg, out);
    mdn_reg<<<1, 256, 0, stream>>>(Wm, Wv, Wg, out);
}